// MKMD_9414568313386
// MI455X (gfx1250) — compile-verified
//
#include <hip/hip_runtime.h>

typedef __attribute__((ext_vector_type(2))) float v2f;
typedef __attribute__((ext_vector_type(8))) float v8f;

#define XD 3
#define HD 3
#define ND 12

__device__ __forceinline__ float swap16(float x) {
    // ds_swizzle group-of-32: and=0x1f, or=0, xor=0x10 -> lane n <-> lane n^16
    int i = __builtin_amdgcn_ds_swizzle(__float_as_int(x), 0x401F);
    return __int_as_float(i);
}

__device__ __forceinline__ float fsig(float x) {
    // sigmoid(x) = 1/(1+exp(-x)); v_exp_f32 is exp2
    float e = __builtin_amdgcn_exp2f(-1.4426950408889634f * x);
    return __builtin_amdgcn_rcpf(1.0f + e);
}

__device__ __forceinline__ float ftanh(float x) {
    float e = __builtin_amdgcn_exp2f(2.8853900817779268f * x); // exp(2x)
    return (e - 1.0f) * __builtin_amdgcn_rcpf(e + 1.0f);
}

// Combined 16x12 step matrix W: rows 0-5 = A rows 0-5 (x_next, h_lin);
// rows 6-8 = s_r, 9-11 = s_z ([W_ih | W_hh | 0 | 0]); rows 12-14 = i_n; row 15 = 0.
__device__ __forceinline__ float wcomb(int r, int c, const float* A,
                                       const float* Wih, const float* Whh) {
    float v = 0.0f;
    if (r < 6) {
        v = A[r * ND + c];
    } else if (r < 12) {
        int wrow = r - 6;                 // 0..5 -> gates r,z rows of W_ih/W_hh
        if (c < XD)            v = Wih[wrow * XD + c];
        else if (c < XD + HD)  v = Whh[wrow * HD + (c - XD)];
    } else if (r < 15) {
        int wrow = 6 + (r - 12);          // n-gate rows of W_ih only
        if (c < XD)            v = Wih[wrow * XD + c];
    }
    return v;
}

__device__ __forceinline__ float bcomb(int r, const float* bih, const float* bhh) {
    if (r >= 6 && r < 12)  return bih[r - 6] + bhh[r - 6];  // s_r, s_z biases
    if (r >= 12 && r < 15) return bih[6 + (r - 12)];        // i_n bias
    return 0.0f;
}

__global__ __launch_bounds__(256) void gru_dict_wmma(
    const float* __restrict__ x0, const float* __restrict__ h0,
    const float* __restrict__ A,  const float* __restrict__ Wih,
    const float* __restrict__ Whh, const float* __restrict__ bih,
    const float* __restrict__ bhh, const int* __restrict__ Kp,
    float* __restrict__ out, int B)
{
    const int K    = Kp[0];
    const int lane = threadIdx.x & 31;
    const int wave = blockIdx.x * (blockDim.x >> 5) + (threadIdx.x >> 5);
    const bool hiH = lane >= 16;
    const int  m   = lane & 15;
    const int  b   = wave * 16 + m;       // this lane's batch element

    // ---- constant A-operand: W as three 16x4 f32 chunks (2 VGPRs each) ----
    // lane m holds cols {k0,k0+1} (low half) or {k0+2,k0+3} (high half)
    const int kb = hiH ? 2 : 0;
    v2f a0, a1, a2;
    a0.x = wcomb(m, 0 + kb,     A, Wih, Whh);
    a0.y = wcomb(m, 0 + kb + 1, A, Wih, Whh);
    a1.x = wcomb(m, 4 + kb,     A, Wih, Whh);
    a1.y = wcomb(m, 4 + kb + 1, A, Wih, Whh);
    a2.x = wcomb(m, 8 + kb,     A, Wih, Whh);
    a2.y = wcomb(m, 8 + kb + 1, A, Wih, Whh);

    // ---- constant bias accumulator in C/D layout (VGPR v = row v + (hi?8:0)) ----
    v8f cb;
    const int rb = hiH ? 8 : 0;
    #pragma unroll
    for (int v = 0; v < 8; ++v) cb[v] = bcomb(rb + v, bih, bhh);

    // ---- uniform scalar weights for h_n = W_hh[6:9]*h + b_hh[6:9] ----
    const float wn00 = Whh[6*HD+0], wn01 = Whh[6*HD+1], wn02 = Whh[6*HD+2];
    const float wn10 = Whh[7*HD+0], wn11 = Whh[7*HD+1], wn12 = Whh[7*HD+2];
    const float wn20 = Whh[8*HD+0], wn21 = Whh[8*HD+1], wn22 = Whh[8*HD+2];
    const float bn0 = bhh[6], bn1 = bhh[7], bn2 = bhh[8];

    // ---- initial state (replicated in both lane halves) ----
    float xs0 = x0[b*XD+0], xs1 = x0[b*XD+1], xs2 = x0[b*XD+2];
    float hs0 = h0[b*HD+0], hs1 = h0[b*HD+1], hs2 = h0[b*HD+2];

    const long plane = (long)B * 3;               // elements per (k) slice
    float* outX = out;
    float* outH = out + (long)K * plane;
    float* outL = out + 2L * (long)K * plane;
    const long boff = (long)b * 3;

    for (int k = 0; k < K; ++k) {
        const float q0 = xs0*xs0, q1 = xs1*xs1, q2 = xs2*xs2;  // x^2
        const float p0 = hs0*hs0, p1 = hs1*hs1, p2 = hs2*hs2;  // h^2

        // B-operand: g = [x(3) h(3) x^2(3) h^2(3)] as three 4x16 chunks.
        // VGPR0 = row k0 (low half) / row k0+2 (high); VGPR1 = k0+1 / k0+3.
        v2f g0, g1, g2;
        g0.x = hiH ? xs2 : xs0;  g0.y = hiH ? hs0 : xs1;  // rows 0,1 / 2,3
        g1.x = hiH ? q0  : hs1;  g1.y = hiH ? q1  : hs2;  // rows 4,5 / 6,7
        g2.x = hiH ? p1  : q2;   g2.y = hiH ? p2  : p0;   // rows 8,9 / 10,11

        v8f d;
        d = __builtin_amdgcn_wmma_f32_16x16x4_f32(false, a0, false, g0, (short)0, cb, false, false);
        d = __builtin_amdgcn_wmma_f32_16x16x4_f32(false, a1, false, g1, (short)0, d,  false, false);
        d = __builtin_amdgcn_wmma_f32_16x16x4_f32(false, a2, false, g2, (short)0, d,  false, false);

        // Gather all 16 D-rows of batch b into every lane (halves swap via LDS permute)
        float r_[16];
        #pragma unroll
        for (int v = 0; v < 8; ++v) {
            const float own = d[v];
            const float oth = swap16(own);
            r_[v]     = hiH ? oth : own;   // rows 0..7
            r_[v + 8] = hiH ? own : oth;   // rows 8..15
        }

        // GRU gates (redundant across halves; stores are half-predicated)
        const float rr0 = fsig(r_[6]),  rr1 = fsig(r_[7]),  rr2 = fsig(r_[8]);
        const float zz0 = fsig(r_[9]),  zz1 = fsig(r_[10]), zz2 = fsig(r_[11]);
        const float hn0 = fmaf(wn00, hs0, fmaf(wn01, hs1, fmaf(wn02, hs2, bn0)));
        const float hn1 = fmaf(wn10, hs0, fmaf(wn11, hs1, fmaf(wn12, hs2, bn1)));
        const float hn2 = fmaf(wn20, hs0, fmaf(wn21, hs1, fmaf(wn22, hs2, bn2)));
        const float nn0 = ftanh(fmaf(rr0, hn0, r_[12]));
        const float nn1 = ftanh(fmaf(rr1, hn1, r_[13]));
        const float nn2 = ftanh(fmaf(rr2, hn2, r_[14]));
        const float hN0 = fmaf(1.0f - zz0, nn0, zz0 * hs0);
        const float hN1 = fmaf(1.0f - zz1, nn1, zz1 * hs1);
        const float hN2 = fmaf(1.0f - zz2, nn2, zz2 * hs2);

        const long o = (long)k * plane + boff;
        if (!hiH) {  // low half: x_next + h_lin
            __builtin_nontemporal_store(r_[0], outX + o + 0);
            __builtin_nontemporal_store(r_[1], outX + o + 1);
            __builtin_nontemporal_store(r_[2], outX + o + 2);
            __builtin_nontemporal_store(r_[3], outL + o + 0);
            __builtin_nontemporal_store(r_[4], outL + o + 1);
            __builtin_nontemporal_store(r_[5], outL + o + 2);
        } else {     // high half: h_next
            __builtin_nontemporal_store(hN0, outH + o + 0);
            __builtin_nontemporal_store(hN1, outH + o + 1);
            __builtin_nontemporal_store(hN2, outH + o + 2);
        }

        xs0 = r_[0]; xs1 = r_[1]; xs2 = r_[2];
        hs0 = hN0;   hs1 = hN1;   hs2 = hN2;
    }
}

extern "C" void kernel_launch(void* const* d_in, const int* in_sizes, int n_in,
                              void* d_out, int out_size, void* d_ws, size_t ws_size,
                              hipStream_t stream) {
    const float* x0  = (const float*)d_in[0];
    const float* h0  = (const float*)d_in[1];
    const float* A   = (const float*)d_in[2];
    const float* Wih = (const float*)d_in[3];
    const float* Whh = (const float*)d_in[4];
    const float* bih = (const float*)d_in[5];
    const float* bhh = (const float*)d_in[6];
    const int*   Kp  = (const int*)d_in[7];
    float* out = (float*)d_out;

    const int B = in_sizes[0] / XD;              // 8192
    // 16 batch elements per wave32, 8 waves per 256-thread block -> 128/block
    const int blocks = B / 128;
    hipLaunchKernelGGL(gru_dict_wmma, dim3(blocks), dim3(256), 0, stream,
                       x0, h0, A, Wih, Whh, bih, bhh, Kp, out, B);
}